// GP_SE_R_INT_19009525252314
// MI455X (gfx1250) — compile-verified
//
#include <hip/hip_runtime.h>
#include <hip/hip_bf16.h>

typedef float v2f __attribute__((ext_vector_type(2)));
typedef float v8f __attribute__((ext_vector_type(8)));

#define NQ 33
#define NT 256
#define NP (NT*NQ)          // 8448 quadrature points
#define LDA 257             // LDS row stride (pad to kill bank conflicts)

// ---- CDNA5 async global->LDS copy (ASYNCcnt path), b32 keeps 4B alignment
__device__ __forceinline__ void async_g2l_b32(uint32_t lds_off, const float* base,
                                              uint32_t byte_off)
{
    asm volatile("global_load_async_to_lds_b32 %0, %1, %2"
                 :: "v"(lds_off), "v"(byte_off), "s"(base)
                 : "memory");
}
__device__ __forceinline__ void wait_async0()
{
    asm volatile("s_wait_asynccnt 0" ::: "memory");
}

// ---------------------------------------------------------------- MLP kernel
__global__ void k_mlp(const float* __restrict__ xt,
                      const float* __restrict__ W1, const float* __restrict__ b1,
                      const float* __restrict__ W2, const float* __restrict__ b2,
                      const float* __restrict__ W3, const float* __restrict__ b3,
                      float* __restrict__ xf, float* __restrict__ uu,
                      float* __restrict__ vv, float* __restrict__ wt)
{
    int p = blockIdx.x * blockDim.x + threadIdx.x;
    if (p >= NP) return;
    int i = p / NQ;
    int k = p - i * NQ;
    float a = xt[2*i + 0];
    float b = xt[2*i + 1];
    float t = (float)k * (1.0f / 32.0f);
    float x = a + (b - a) * t;
    float wk = (k == 0 || k == NQ-1) ? 1.0f : ((k & 1) ? 4.0f : 2.0f);
    wk *= (1.0f / 32.0f) / 3.0f;
    float wq = (b - a) * wk;

    float h2[10];
#pragma unroll
    for (int q = 0; q < 10; ++q) h2[q] = b2[q];
    for (int j = 0; j < 60; ++j) {
        float h1 = tanhf(x * W1[j] + b1[j]);
#pragma unroll
        for (int q = 0; q < 10; ++q) h2[q] += h1 * W2[j*10 + q];
    }
    float s = b3[0];
#pragma unroll
    for (int q = 0; q < 10; ++q) s += tanhf(h2[q]) * W3[q];
    float bv = 1.0f / (1.0f + __expf(-s));

    xf[p] = x;
    wt[p] = wq;
    vv[p] = wq * bv;
    uu[p] = wq * (1.0f - bv * bv);
}

// --------------------------------------------------------- M (upper triangle)
__device__ __forceinline__ int tri_start(int r) { return r*NT - (r*(r-1))/2; }

__global__ void k_M(const float* __restrict__ xf, const float* __restrict__ uu,
                    const float* __restrict__ vv, const float* __restrict__ wt,
                    const float* __restrict__ sf, const float* __restrict__ ls,
                    float* __restrict__ M)
{
    int lane = threadIdx.x & 31;
    int wave = threadIdx.x >> 5;
    int pid  = blockIdx.x * 8 + wave;
    const int NPAIR = NT*(NT+1)/2;          // 32896 upper-tri pairs
    if (pid >= NPAIR) return;               // wave-uniform exit

    // invert triangular index: pid -> (i,j), j >= i
    float fr = (513.0f - sqrtf(fmaxf(0.0f, 513.0f*513.0f - 8.0f*(float)pid))) * 0.5f;
    int i = (int)fr;
    if (i < 0) i = 0;
    if (i > NT-1) i = NT-1;
    while (i < NT-1 && tri_start(i+1) <= pid) ++i;
    while (i > 0 && tri_start(i) > pid) --i;
    int j = i + (pid - tri_start(i));

    float sigf = sf[0];
    float l    = ls[0];
    float c2   = -0.5f / (l * l);
    int bi = i * NQ, bj = j * NQ;

    // lane owns row k = lane (i-side values live in registers);
    // j-side operands are wave-uniform -> scalar loads in the hot loop.
    float xi = xf[bi + lane];
    float ui = uu[bi + lane];
    float vi = vv[bi + lane];

    float acc = 0.0f;
    for (int q = 0; q < NQ; ++q) {          // 33 iterations, uniform operands
        float xq = xf[bj + q];
        float wq = wt[bj + q];
        float vq = vv[bj + q];
        float d  = xi - xq;
        acc += (ui * wq + vi * vq) * __expf(d * d * c2);
    }
    // row k = 32: q = lane covers 0..31
    {
        float xi32 = xf[bi + 32];
        float ui32 = uu[bi + 32];
        float vi32 = vv[bi + 32];
        float xq = xf[bj + lane];
        float wq = wt[bj + lane];
        float vq = vv[bj + lane];
        float d  = xi32 - xq;
        acc += (ui32 * wq + vi32 * vq) * __expf(d * d * c2);
        // final corner term (k=32, q=32) on lane 0
        if (lane == 0) {
            float xq32 = xf[bj + 32];
            float d2   = xi32 - xq32;
            acc += (ui32 * wt[bj + 32] + vi32 * vv[bj + 32]) * __expf(d2 * d2 * c2);
        }
    }
#pragma unroll
    for (int off = 16; off; off >>= 1) acc += __shfl_xor(acc, off, 32);
    if (lane == 0) M[i*NT + j] = sigf * sigf * acc;
}

// ----------------------------------------------------------------- build kyy
__global__ void k_kyy(const float* __restrict__ M, const float* __restrict__ sn,
                      float* __restrict__ kyy)
{
    int i = blockIdx.x, j = threadIdx.x;
    float m = (j >= i) ? M[i*NT + j] : M[j*NT + i];
    if (i == j) m += sn[0] * sn[0];
    kyy[i*NT + j] = m;
}

// --------------------------------------- min-eig estimate via power iteration
__global__ void __launch_bounds__(256) k_eigmin(const float* __restrict__ kyy,
                                                float* __restrict__ mine)
{
    __shared__ float As[NT * LDA];          // matrix resident in LDS
    __shared__ float xv[NT], yv[NT], red[NT];
    int tid = threadIdx.x;

    // async global->LDS staging of kyy (ASYNCcnt path)
    uint32_t lbase = (uint32_t)(uintptr_t)&As[0];
#pragma unroll 1
    for (int t = 0; t < NT; ++t) {
        int e = tid + t * NT;
        int r = e >> 8, c = e & 255;
        async_g2l_b32(lbase + (uint32_t)(r*LDA + c) * 4u, kyy, (uint32_t)e * 4u);
    }
    wait_async0();
    __syncthreads();

    xv[tid] = 1.0f / 16.0f;                 // unit-norm constant start vector
    __syncthreads();
    float lam = 0.0f;
    for (int it = 0; it < 128; ++it) {
        float acc = 0.0f;
        for (int j = 0; j < NT; ++j) acc += As[tid*LDA + j] * xv[j];
        yv[tid] = acc; red[tid] = acc * acc;
        __syncthreads();
        for (int s = 128; s; s >>= 1) { if (tid < s) red[tid] += red[tid + s]; __syncthreads(); }
        float nrm = sqrtf(red[0]);
        lam = nrm;
        xv[tid] = yv[tid] / fmaxf(nrm, 1e-30f);
        __syncthreads();
    }
    float shift = lam * 1.0001f + 1e-3f;    // B = shift*I - A is PD
    xv[tid] = 1.0f / 16.0f;
    __syncthreads();
    float nu = 0.0f;
    for (int it = 0; it < 128; ++it) {
        float acc = 0.0f;
        for (int j = 0; j < NT; ++j) acc += As[tid*LDA + j] * xv[j];
        acc = shift * xv[tid] - acc;
        yv[tid] = acc; red[tid] = acc * acc;
        __syncthreads();
        for (int s = 128; s; s >>= 1) { if (tid < s) red[tid] += red[tid + s]; __syncthreads(); }
        float nrm = sqrtf(red[0]);
        nu = nrm;
        xv[tid] = yv[tid] / fmaxf(nrm, 1e-30f);
        __syncthreads();
    }
    if (tid == 0) mine[0] = shift - nu;     // lambda_min estimate
}

// ------------------------------------------------------- conditional jitter
__global__ void k_corr(float* __restrict__ kyy, const float* __restrict__ mine)
{
    int i = blockIdx.x, j = threadIdx.x;
    float mn = mine[0];
    if (mn < 1e-6f) {
        float d = (i == j) ? mn : 0.0f;
        kyy[i*NT + j] += 1.1f * fabsf(1e-6f - d);
    }
}

// ------------- in-LDS blocked Cholesky (WMMA f32 trailing updates) + solves
__global__ void __launch_bounds__(256) k_chol(const float* __restrict__ kyy,
                                              const float* __restrict__ y,
                                              float* __restrict__ out)
{
    __shared__ float As[NT * LDA];          // 257 KB < 320 KB/WGP LDS
    __shared__ float zv[NT];
    __shared__ float vres[NT];
    int tid = threadIdx.x;
    int lane = tid & 31, wave = tid >> 5;
    int m = lane & 15, hf = lane >> 4;      // WMMA lane decomposition

    // async global->LDS staging of kyy (ASYNCcnt path)
    uint32_t lbase = (uint32_t)(uintptr_t)&As[0];
#pragma unroll 1
    for (int t = 0; t < NT; ++t) {
        int e = tid + t * NT;
        int r = e >> 8, c = e & 255;
        async_g2l_b32(lbase + (uint32_t)(r*LDA + c) * 4u, kyy, (uint32_t)e * 4u);
    }
    wait_async0();
    __syncthreads();

    for (int kb = 0; kb < 16; ++kb) {
        int p0 = kb * 16;
        // ---- panel factorization (16 columns, full height)
        for (int c = 0; c < 16; ++c) {
            int jc = p0 + c;
            if (tid == 0) As[jc*LDA + jc] = sqrtf(As[jc*LDA + jc]);
            __syncthreads();
            if (tid > jc) As[tid*LDA + jc] /= As[jc*LDA + jc];
            __syncthreads();
            for (int j2 = jc + 1; j2 < p0 + 16; ++j2)
                if (tid >= j2) As[tid*LDA + j2] -= As[tid*LDA + jc] * As[j2*LDA + jc];
            __syncthreads();
        }
        // ---- trailing SYRK update: A[ib,jb] -= P_ib * P_jb^T  (WMMA f32)
        int cnt = 0;
        for (int jb = kb + 1; jb < 16; ++jb) {
            for (int ib = jb; ib < 16; ++ib) {
                if ((cnt & 7) == wave) {     // wave-uniform; EXEC stays all-1s
                    int rA = ib * 16, rB = jb * 16;
                    v8f acc = {0.f, 0.f, 0.f, 0.f, 0.f, 0.f, 0.f, 0.f};
#pragma unroll
                    for (int k0 = 0; k0 < 16; k0 += 4) {
                        int kc = p0 + k0 + hf * 2;
                        v2f a, b;
                        a.x = As[(rA + m)*LDA + kc];   // A 16x4: lanes 0-15 K=0,1; 16-31 K=2,3
                        a.y = As[(rA + m)*LDA + kc + 1];
                        b.x = As[(rB + m)*LDA + kc];   // B 4x16 = P_jb^T
                        b.y = As[(rB + m)*LDA + kc + 1];
                        acc = __builtin_amdgcn_wmma_f32_16x16x4_f32(
                                  false, a, false, b, (short)0, acc, false, false);
                    }
#pragma unroll
                    for (int r = 0; r < 8; ++r)
                        As[(rA + hf*8 + r)*LDA + rB + m] -= acc[r];
                }
                ++cnt;
            }
        }
        __syncthreads();
    }

    // ---- forward solve L z = y
    zv[tid] = y[tid];
    __syncthreads();
    for (int j = 0; j < NT; ++j) {
        if (tid == j) zv[j] = zv[j] / As[j*LDA + j];
        __syncthreads();
        if (tid > j) zv[tid] -= As[tid*LDA + j] * zv[j];
        __syncthreads();
    }
    // ---- backward solve L^T v = z
    for (int j = NT - 1; j >= 0; --j) {
        if (tid == j) vres[j] = zv[j] / As[j*LDA + j];
        __syncthreads();
        if (tid < j) zv[tid] -= As[j*LDA + tid] * vres[j];
        __syncthreads();
    }
    // ---- outputs: c = L^T (upper, zeros below), then v
    for (int e = tid; e < NT*NT; e += 256) {
        int i = e >> 8, j = e & 255;
        out[e] = (i <= j) ? As[j*LDA + i] : 0.0f;
    }
    out[NT*NT + tid] = vres[tid];
}

// ---------------------------------------------------------------- launcher
extern "C" void kernel_launch(void* const* d_in, const int* in_sizes, int n_in,
                              void* d_out, int out_size, void* d_ws, size_t ws_size,
                              hipStream_t stream)
{
    const float* xt = (const float*)d_in[0];
    const float* yt = (const float*)d_in[1];
    const float* sf = (const float*)d_in[2];
    const float* ls = (const float*)d_in[3];
    const float* sn = (const float*)d_in[4];
    const float* W1 = (const float*)d_in[5];
    const float* b1 = (const float*)d_in[6];
    const float* W2 = (const float*)d_in[7];
    const float* b2 = (const float*)d_in[8];
    const float* W3 = (const float*)d_in[9];
    const float* b3 = (const float*)d_in[10];
    (void)in_sizes; (void)n_in; (void)out_size; (void)ws_size;

    float* ws   = (float*)d_ws;
    float* xf   = ws;                 // 8448
    float* uu   = xf + NP;            // 8448
    float* vv   = uu + NP;            // 8448
    float* wt   = vv + NP;            // 8448
    float* M    = wt + NP;            // 65536
    float* kyy  = M  + NT*NT;         // 65536
    float* mine = kyy + NT*NT;        // 1
    float* out  = (float*)d_out;

    k_mlp<<<(NP + 255)/256, 256, 0, stream>>>(xt, W1, b1, W2, b2, W3, b3, xf, uu, vv, wt);
    int npair = NT*(NT+1)/2;
    k_M<<<(npair + 7)/8, 256, 0, stream>>>(xf, uu, vv, wt, sf, ls, M);
    k_kyy<<<NT, NT, 0, stream>>>(M, sn, kyy);
    k_eigmin<<<1, NT, 0, stream>>>(kyy, mine);
    k_corr<<<NT, NT, 0, stream>>>(kyy, mine);
    k_chol<<<1, NT, 0, stream>>>(kyy, yt, out);
}